// Set2Set_17093969838317
// MI455X (gfx1250) — compile-verified
//
#include <hip/hip_runtime.h>

// Set2Set (D=128, B=2048 graphs x 256 nodes, 6 iterations) for gfx1250.
//   lstm_step: gates = [q_star|h] @ [W_ih|W_hh]^T + b via v_wmma_f32_16x16x4_f32,
//              software-pipelined weight loads (double buffer), split K regions.
//   attn_step: per-graph block DMAs its 128KB feat tile to LDS with
//              global_load_async_to_lds_b128, computes e = feat@q, segment
//              softmax, readout, writes q_star[:, 128:].

typedef float v2f __attribute__((ext_vector_type(2)));
typedef float v8f __attribute__((ext_vector_type(8)));

#define DIMD   128
#define NGRAPH 2048
#define NODES  256          // nodes per graph
#define KTOT   384          // 256 (q_star) + 128 (h)
#define XSTR   388          // padded LDS stride for X tile
#define GSTR   516          // padded LDS stride for gate tile
#define TSTR   132          // padded LDS row stride (floats) for feat tile

__device__ __forceinline__ float sigf(float x) {
    return 1.0f / (1.0f + __expf(-x));
}

__device__ __forceinline__ v2f ld2(const float* p) {
    return *(const v2f*)p;   // 8B-aligned by construction (even element offsets)
}

__global__ void init_zero(float* __restrict__ qstar,
                          float* __restrict__ h, float* __restrict__ c) {
    int i = blockIdx.x * 256 + threadIdx.x;      // grid = 2048*256 = 524288
    qstar[i] = 0.0f;
    if (i < NGRAPH * DIMD) { h[i] = 0.0f; c[i] = 0.0f; }
}

// ---------------------------------------------------------------------------
// LSTM step: 128 blocks x 256 threads. Block = 16 graphs; 8 waves x 64 gates.
// ---------------------------------------------------------------------------
__global__ void lstm_step(float* __restrict__ qstar,
                          float* __restrict__ h, float* __restrict__ c,
                          const float* __restrict__ W_ih,
                          const float* __restrict__ W_hh,
                          const float* __restrict__ b_ih,
                          const float* __restrict__ b_hh) {
    __shared__ float X[16][XSTR];   // [graph-in-block][k]  (k<256: q_star, else h)
    __shared__ float G[16][GSTR];   // gates

    const int tid = threadIdx.x;
    const int g0  = blockIdx.x * 16;

    // Stage X tile (16 x 384) into LDS, coalesced.
    for (int idx = tid; idx < 16 * KTOT; idx += 256) {
        int m = idx / KTOT, k = idx - m * KTOT;
        X[m][k] = (k < 256) ? qstar[(g0 + m) * 256 + k]
                            : h[(g0 + m) * DIMD + (k - 256)];
    }
    __syncthreads();

    const int lane  = tid & 31;
    const int wv    = tid >> 5;           // wave id 0..7
    const int mlo   = lane & 15;          // A: M row / B: N col within tile
    const int khi   = (lane >> 4) * 2;    // K sub-offset per ISA f32 layout
    const int nbase = wv * 64;

    v8f acc[4];
    for (int t = 0; t < 4; ++t) acc[t] = (v8f){0,0,0,0,0,0,0,0};

    int ncol[4];
    const float* wih[4];
    const float* whh[4];
    for (int t = 0; t < 4; ++t) {
        ncol[t] = nbase + t * 16 + mlo;
        wih[t]  = W_ih + ncol[t] * 256 + khi;
        whh[t]  = W_hh + ncol[t] * 128 + khi;
    }
    const float* xrow  = &X[mlo][khi];        // q_star region of X row
    const float* xrow2 = &X[mlo][256 + khi];  // h region of X row

    // Software-pipelined GEMM: double-buffered B operands, no conditionals
    // in the steady-state loops (boundary iterations peeled).
    v2f bb[4], nb[4];
    for (int t = 0; t < 4; ++t) bb[t] = ld2(wih[t]);          // k = 0

    for (int k = 0; k < 252; k += 4) {
        v2f a = ld2(xrow + k);
        for (int t = 0; t < 4; ++t) nb[t] = ld2(wih[t] + k + 4);   // prefetch k+4
        for (int t = 0; t < 4; ++t)
            acc[t] = __builtin_amdgcn_wmma_f32_16x16x4_f32(
                false, a, false, bb[t], (short)0, acc[t], false, false);
        for (int t = 0; t < 4; ++t) bb[t] = nb[t];
    }
    {   // k = 252 (last W_ih step), prefetch first W_hh step
        v2f a = ld2(xrow + 252);
        for (int t = 0; t < 4; ++t) nb[t] = ld2(whh[t]);
        for (int t = 0; t < 4; ++t)
            acc[t] = __builtin_amdgcn_wmma_f32_16x16x4_f32(
                false, a, false, bb[t], (short)0, acc[t], false, false);
        for (int t = 0; t < 4; ++t) bb[t] = nb[t];
    }
    for (int k = 0; k < 124; k += 4) {
        v2f a = ld2(xrow2 + k);
        for (int t = 0; t < 4; ++t) nb[t] = ld2(whh[t] + k + 4);
        for (int t = 0; t < 4; ++t)
            acc[t] = __builtin_amdgcn_wmma_f32_16x16x4_f32(
                false, a, false, bb[t], (short)0, acc[t], false, false);
        for (int t = 0; t < 4; ++t) bb[t] = nb[t];
    }
    {   // k = 124 (last W_hh step)
        v2f a = ld2(xrow2 + 124);
        for (int t = 0; t < 4; ++t)
            acc[t] = __builtin_amdgcn_wmma_f32_16x16x4_f32(
                false, a, false, bb[t], (short)0, acc[t], false, false);
    }

    // Add biases (per-column) and spill gates to LDS per the 16x16 f32 C layout:
    // VGPR r -> M = r + 8*(lane>>4); N = lane&15 within tile.
    for (int t = 0; t < 4; ++t) {
        float bias = b_ih[ncol[t]] + b_hh[ncol[t]];
        for (int r = 0; r < 8; ++r) {
            int m = r + (lane >> 4) * 8;
            G[m][ncol[t]] = acc[t][r] + bias;
        }
    }
    __syncthreads();

    // Activations + state update. torch gate order: i, f, g, o.
    const int d    = tid & 127;
    const int half = tid >> 7;
    for (int j = 0; j < 8; ++j) {
        int m = half + 2 * j;
        int g = g0 + m;
        float gi = sigf(G[m][d]);
        float gf = sigf(G[m][128 + d]);
        float gg = tanhf(G[m][256 + d]);
        float go = sigf(G[m][384 + d]);
        float cc = gf * c[g * DIMD + d] + gi * gg;
        c[g * DIMD + d] = cc;
        float hh = go * tanhf(cc);
        h[g * DIMD + d] = hh;
        qstar[g * 256 + d] = hh;          // q (lower half of q_star)
    }
}

// ---------------------------------------------------------------------------
// Attention + readout: 2048 blocks (one graph) x 256 threads.
// feat tile (256x128 f32, row pad 4) DMA'd into LDS via async-to-LDS B128.
// ---------------------------------------------------------------------------
__global__ void attn_step(const float* __restrict__ feat,
                          const float* __restrict__ h,
                          float* __restrict__ qstar) {
    extern __shared__ float smem[];
    float* tile  = smem;                        // NODES * TSTR
    float* qs    = smem + NODES * TSTR;         // 128
    float* alpha = qs + DIMD;                   // 256
    float* sc    = alpha + NODES;               // 256

    const int tid = threadIdx.x;
    const int b   = blockIdx.x;

    // Async DMA: thread t owns node row t; 32 x B128 per row (512 B).
    {
        const float* gbase = feat + (size_t)b * NODES * DIMD;
        unsigned loff = (unsigned)(size_t)(void*)tile + (unsigned)tid * (TSTR * 4);
        unsigned voff = (unsigned)tid * (DIMD * 4);
        for (int j = 0; j < 32; ++j) {
            asm volatile("global_load_async_to_lds_b128 %0, %1, %2"
                         :: "v"(loff), "v"(voff), "s"(gbase) : "memory");
            loff += 16; voff += 16;
        }
    }
    if (tid < DIMD) qs[tid] = h[b * DIMD + tid];   // q = h[b]
    asm volatile("s_wait_asynccnt 0" ::: "memory");
    __syncthreads();

    // e[t] = feat[t,:] . q
    float e = 0.0f;
    {
        const float* row = tile + tid * TSTR;
        for (int d2 = 0; d2 < DIMD; ++d2) e += row[d2] * qs[d2];
    }
    sc[tid] = e;
    __syncthreads();
    // max reduce
    for (int s = 128; s > 0; s >>= 1) {
        if (tid < s) sc[tid] = fmaxf(sc[tid], sc[tid + s]);
        __syncthreads();
    }
    float mx = sc[0];
    __syncthreads();
    float z = __expf(e - mx);
    alpha[tid] = z;
    sc[tid] = z;
    __syncthreads();
    // sum reduce
    for (int s = 128; s > 0; s >>= 1) {
        if (tid < s) sc[tid] += sc[tid + s];
        __syncthreads();
    }
    float inv = 1.0f / sc[0];
    __syncthreads();

    // readout[d] = sum_n alpha[n]/Z * feat[n][d]; split nodes across halves.
    const int d    = tid & 127;
    const int half = tid >> 7;
    float racc = 0.0f;
    for (int n = half * 128; n < half * 128 + 128; ++n)
        racc += alpha[n] * tile[n * TSTR + d];
    sc[half * 128 + d] = racc;
    __syncthreads();
    if (tid < DIMD)
        qstar[b * 256 + 128 + tid] = (sc[tid] + sc[128 + tid]) * inv;
}

extern "C" void kernel_launch(void* const* d_in, const int* in_sizes, int n_in,
                              void* d_out, int out_size, void* d_ws, size_t ws_size,
                              hipStream_t stream) {
    const float* feat = (const float*)d_in[0];
    const float* W_ih = (const float*)d_in[1];
    const float* W_hh = (const float*)d_in[2];
    const float* b_ih = (const float*)d_in[3];
    const float* b_hh = (const float*)d_in[4];
    // d_in[5] segment_ids, d_in[6] num_graphs: segments are uniform (256/graph).

    float* qstar = (float*)d_out;                 // (2048, 256) f32, evolves in place
    float* h = (float*)d_ws;                      // (2048, 128)
    float* c = h + NGRAPH * DIMD;                 // (2048, 128)

    init_zero<<<NGRAPH, 256, 0, stream>>>(qstar, h, c);

    const size_t smem = (NODES * TSTR + DIMD + NODES + NODES) * sizeof(float);
    for (int it = 0; it < 6; ++it) {
        lstm_step<<<NGRAPH / 16, 256, 0, stream>>>(qstar, h, c, W_ih, W_hh, b_ih, b_hh);
        attn_step<<<NGRAPH, 256, smem, stream>>>(feat, h, qstar);
    }
}